// AggregatedContrastiveLoss_32057635897734
// MI455X (gfx1250) — compile-verified
//
#include <hip/hip_runtime.h>
#include <math.h>

#define D        128
#define C        150
#define INV_T    5.0f       // 1 / TEMPERATURE
#define CPAD     160        // 150 padded to 10 tiles of 16

// workspace layout (in floats)
#define OFF_ACC  0                       // [2*C*D]  group sums
#define OFF_CNT  (2*C*D)                 // [2*C]    group counts
#define OFF_A    (2*C*D + 2*C + 4)       // [CPAD*D] normalized A means (rounded to /4)
#define OFF_B    (OFF_A + CPAD*D)        // [CPAD*D] normalized B means
#define OFF_L    (OFF_B + CPAD*D)        // [CPAD*CPAD] logits

#define NB       256
#define TPB      256
#define UNROLL   8

typedef float v2f __attribute__((ext_vector_type(2)));
typedef float v4f __attribute__((ext_vector_type(4)));
typedef float v8f __attribute__((ext_vector_type(8)));

__global__ void k_zero(float* ws, int n) {
    int i = blockIdx.x * blockDim.x + threadIdx.x;
    if (i < n) ws[i] = 0.0f;
}

// One wave per point: lane l holds dims [4l, 4l+4) of the 128-dim row (one b128 NT load),
// accumulates into LDS-privatized [2][C][D] bins, block-level flush via global fp32 atomics.
__global__ void k_accum(const float* __restrict__ pred,
                        const int*   __restrict__ valid,
                        const int*   __restrict__ seg,
                        const int*   __restrict__ grp,
                        float* __restrict__ accum,   // [2*C*D]
                        float* __restrict__ cnt,     // [2*C]
                        int n)
{
    extern __shared__ float lds[];                 // 2*C*D + 2*C floats
    float* lcnt = lds + 2 * C * D;
    for (int i = threadIdx.x; i < 2 * C * D + 2 * C; i += blockDim.x) lds[i] = 0.0f;
    __syncthreads();

    const int lane   = threadIdx.x & 31;
    const int wavesB = blockDim.x >> 5;
    const int gwave  = blockIdx.x * wavesB + (threadIdx.x >> 5);
    const int nwaves = gridDim.x * wavesB;

    for (int base = gwave * UNROLL; base < n; base += nwaves * UNROLL) {
        v4f row[UNROLL];
        int sc[UNROLL], gg[UNROLL], ok[UNROLL];
        #pragma unroll
        for (int u = 0; u < UNROLL; ++u) {
            int p = base + u;
            if (p < n) {
                const v4f* rp = (const v4f*)(pred + (size_t)p * D) + lane;
                row[u] = __builtin_nontemporal_load(rp);     // no reuse: TH=NT
                int s = seg[p];
                ok[u] = (valid[p] > 0) & (s != -1);
                sc[u] = min(max(s, 0), C - 1);
                gg[u] = grp[p];
            } else { ok[u] = 0; sc[u] = 0; gg[u] = 0; row[u] = (v4f)0.0f; }
        }
        #pragma unroll
        for (int u = 0; u < UNROLL; ++u) {
            if (ok[u]) {
                float* dst = lds + (size_t)(gg[u] * C + sc[u]) * D + lane * 4;
                atomicAdd(dst + 0, row[u].x);
                atomicAdd(dst + 1, row[u].y);
                atomicAdd(dst + 2, row[u].z);
                atomicAdd(dst + 3, row[u].w);
                if (lane == 0) atomicAdd(&lcnt[gg[u] * C + sc[u]], 1.0f);
            }
        }
    }
    __syncthreads();
    for (int i = threadIdx.x; i < 2 * C * D; i += blockDim.x) {
        float v = lds[i];
        if (v != 0.0f) atomicAdd(&accum[i], v);
    }
    for (int i = threadIdx.x; i < 2 * C; i += blockDim.x) {
        float v = lcnt[i];
        if (v != 0.0f) atomicAdd(&cnt[i], v);
    }
}

// One block (128 threads) per (group, class): mean then L2-normalize the 128-dim row.
__global__ void k_norm(const float* __restrict__ accum, const float* __restrict__ cnt,
                       float* __restrict__ aB, float* __restrict__ bB)
{
    int b = blockIdx.x;              // 0 .. 2*C-1
    int g = b / C, c = b % C;
    int d = threadIdx.x;             // 0 .. 127
    float m = accum[(g * C + c) * D + d] / fmaxf(cnt[g * C + c], 1.0f);
    float sq = m * m;
    #pragma unroll
    for (int off = 16; off > 0; off >>= 1) sq += __shfl_down(sq, off, 32);
    __shared__ float red[4];
    if ((threadIdx.x & 31) == 0) red[threadIdx.x >> 5] = sq;
    __syncthreads();
    float tot = red[0] + red[1] + red[2] + red[3];
    float* out = (g == 0) ? aB : bB;
    out[c * D + d] = m * (1.0f / sqrtf(tot));
}

// logits[CPAD][CPAD] = (a @ b^T) * INV_T via V_WMMA_F32_16X16X4_F32.
// One wave per 16x16 tile; K-loop steps of 4.
// f32 A 16x4 layout (ISA 7.12.2): lane L: M = L&15, VGPRj holds K = (L>>4)*2 + j.
// f32 B 4x16: lane L: N = L&15, VGPRj holds K = (L>>4)*2 + j.  B[k][n] = b[n][k].
// f32 C/D 16x16: VGPRr: lanes0-15 M=r, lanes16-31 M=r+8; N = L&15.
__global__ void k_gemm(const float* __restrict__ aB, const float* __restrict__ bB,
                       float* __restrict__ logits)
{
    const int lane  = threadIdx.x & 31;
    const int wave  = (blockIdx.x * blockDim.x + threadIdx.x) >> 5;
    if (wave >= (CPAD / 16) * (CPAD / 16)) return;   // uniform per wave
    const int tm    = wave / (CPAD / 16);
    const int tn    = wave % (CPAD / 16);
    const int half  = lane >> 4;
    const int i16   = lane & 15;
    const float* ap = aB + (tm * 16 + i16) * D;
    const float* bp = bB + (tn * 16 + i16) * D;

    v8f acc = {};
    #pragma unroll 4
    for (int k0 = 0; k0 < D; k0 += 4) {
        int kk = k0 + half * 2;
        v2f av; av.x = ap[kk]; av.y = ap[kk + 1];
        v2f bv; bv.x = bp[kk]; bv.y = bp[kk + 1];
        acc = __builtin_amdgcn_wmma_f32_16x16x4_f32(
            /*neg_a=*/false, av, /*neg_b=*/false, bv,
            /*c_mod=*/(short)0, acc, /*reuse_a=*/false, /*reuse_b=*/false);
    }
    #pragma unroll
    for (int r = 0; r < 8; ++r) {
        int rowi = tm * 16 + r + half * 8;
        int coli = tn * 16 + i16;
        logits[rowi * CPAD + coli] = acc[r] * INV_T;
    }
}

// Symmetric InfoNCE over the 150x150 valid region.
__global__ void k_loss(const float* __restrict__ logits, float* __restrict__ out)
{
    __shared__ float lse_r[C], lse_c[C], dia[C];
    int t = threadIdx.x;
    if (t < C) {
        float mx = -1e30f, s = 0.0f;
        for (int j = 0; j < C; ++j) mx = fmaxf(mx, logits[t * CPAD + j]);
        for (int j = 0; j < C; ++j) s += expf(logits[t * CPAD + j] - mx);
        lse_r[t] = mx + logf(s);
        mx = -1e30f; s = 0.0f;
        for (int j = 0; j < C; ++j) mx = fmaxf(mx, logits[j * CPAD + t]);
        for (int j = 0; j < C; ++j) s += expf(logits[j * CPAD + t] - mx);
        lse_c[t] = mx + logf(s);
        dia[t] = logits[t * CPAD + t];
    }
    __syncthreads();
    if (t == 0) {
        float la = 0.0f, lb = 0.0f;
        for (int i = 0; i < C; ++i) { la += lse_r[i] - dia[i]; lb += lse_c[i] - dia[i]; }
        out[0] = 0.5f * (la / C + lb / C);   // LOSS_WEIGHT = 1
    }
}

extern "C" void kernel_launch(void* const* d_in, const int* in_sizes, int n_in,
                              void* d_out, int out_size, void* d_ws, size_t ws_size,
                              hipStream_t stream) {
    const float* pred  = (const float*)d_in[0];
    // d_in[1] = target (unused by the loss math)
    const int*   valid = (const int*)d_in[2];
    const int*   seg   = (const int*)d_in[3];
    const int*   grp   = (const int*)d_in[4];
    const int    n     = in_sizes[2];

    float* ws     = (float*)d_ws;
    float* accum  = ws + OFF_ACC;
    float* cnt    = ws + OFF_CNT;
    float* aB     = ws + OFF_A;
    float* bB     = ws + OFF_B;
    float* logits = ws + OFF_L;

    const int zeroN = OFF_L;   // accum + counts + padded a/b buffers
    k_zero<<<(zeroN + 255) / 256, 256, 0, stream>>>(ws, zeroN);

    const size_t ldsBytes = (size_t)(2 * C * D + 2 * C) * sizeof(float);
    k_accum<<<NB, TPB, ldsBytes, stream>>>(pred, valid, seg, grp, accum, cnt, n);

    k_norm<<<2 * C, D, 0, stream>>>(accum, cnt, aB, bB);

    k_gemm<<<(CPAD / 16) * (CPAD / 16) / 4, 128, 0, stream>>>(aB, bB, logits);

    k_loss<<<1, 256, 0, stream>>>(logits, (float*)d_out);
}